// VladPooling_78065325572148
// MI455X (gfx1250) — compile-verified
//
#include <hip/hip_runtime.h>
#include <math.h>

// ---------------------------------------------------------------------------
// VLAD pooling for MI455X (gfx1250, wave32).
// features (32,512,8,100) f32, cluster_score (32,10,8,100) f32,
// centroids (10,512) f32 -> out (32, 8*512) f32.
//
// Roofline: features = 52.4 MB read exactly once -> ~2.3 us @ 23.3 TB/s HBM;
// FLOPs ~0.26 G (fp32) -> bandwidth-bound. fp32 precision is required
// (softmax + L2 norm), so the matrix path is V_WMMA_F32_16X16X4_F32.
// Data movement uses GLOBAL_LOAD_ASYNC_TO_LDS_B32 (ASYNCcnt-tracked,
// double-buffered) so the feature stream never round-trips through VGPRs.
// ---------------------------------------------------------------------------

typedef __attribute__((ext_vector_type(2))) float v2f;
typedef __attribute__((ext_vector_type(8))) float v8f;

typedef __attribute__((address_space(1))) int as1_int;   // global
typedef __attribute__((address_space(3))) int as3_int;   // LDS

#define VB   32      // batch
#define VC   512     // channels
#define VHW  800     // H*W
#define VKG  10      // clusters (gather)
#define VKC  8       // clusters kept
#define CTILES (VC / 16)

// ---------------- CDNA5 async copy: global -> LDS (ASYNCcnt) ----------------
__device__ __forceinline__ void async_copy_f32(float* lds_dst, const float* g_src) {
#if defined(__has_builtin) && __has_builtin(__builtin_amdgcn_global_load_async_to_lds_b32)
    __builtin_amdgcn_global_load_async_to_lds_b32(
        (as1_int*)(size_t)g_src,          // shed const + retype via int round-trip
        (as3_int*)lds_dst,                // generic -> LDS addrspacecast
        /*imm offset=*/0, /*cpol=*/0);
#else
    unsigned loff = (unsigned)(size_t)(as3_int*)lds_dst;
    asm volatile("global_load_async_to_lds_b32 %0, %1, off"
                 :: "v"(loff), "v"(g_src) : "memory");
#endif
}

template <int N>
__device__ __forceinline__ void wait_asynccnt() {
#if defined(__has_builtin) && __has_builtin(__builtin_amdgcn_s_wait_asynccnt)
    __builtin_amdgcn_s_wait_asynccnt(N);
#else
    asm volatile("s_wait_asynccnt %0" :: "i"(N) : "memory");
#endif
    asm volatile("" ::: "memory");   // keep LDS reads below the wait
}

// ---------------------------------------------------------------------------
// Kernel 1: softmax over the BATCH axis (reference uses axis=0).
// One thread per (k,hw) position; loads coalesced across positions.
// ---------------------------------------------------------------------------
__global__ void vlad_softmax_kernel(const float* __restrict__ cs,
                                    float* __restrict__ sa) {
    int p = blockIdx.x * blockDim.x + threadIdx.x;   // (k,hw) flat
    if (p >= VKG * VHW) return;
    float x[VB];
    float m = -INFINITY;
#pragma unroll
    for (int b = 0; b < VB; ++b) {
        x[b] = cs[(size_t)b * (VKG * VHW) + p];
        m = fmaxf(m, x[b]);
    }
    float s = 0.f;
#pragma unroll
    for (int b = 0; b < VB; ++b) {
        x[b] = expf(x[b] - m);
        s += x[b];
    }
    float inv = 1.f / s;
#pragma unroll
    for (int b = 0; b < VB; ++b) {
        sa[(size_t)b * (VKG * VHW) + p] = x[b] * inv;
    }
}

// ---------------------------------------------------------------------------
// Kernel 2: a_sum[b,k] = sum_hw soft_assign[b,k,hw].  One wave32 per (b,k).
// ---------------------------------------------------------------------------
__global__ void __launch_bounds__(32) vlad_asum_kernel(const float* __restrict__ sa,
                                                       float* __restrict__ asum) {
    int bk   = blockIdx.x;          // 0 .. B*KG-1
    int lane = threadIdx.x;         // 0 .. 31
    const float* row = sa + (size_t)bk * VHW;
    float s = 0.f;
    for (int i = lane; i < VHW; i += 32) s += row[i];
#pragma unroll
    for (int off = 16; off > 0; off >>= 1) s += __shfl_xor(s, off, 32);
    if (lane == 0) asum[bk] = s;
}

// ---------------------------------------------------------------------------
// Kernel 3: WMMA GEMM + centroid subtraction.
// One wave per (batch, 16-channel tile):
//   D[16k x 16c] += A[16k x 4hw] x B[4hw x 16c]  (v_wmma_f32_16x16x4_f32)
// Both operands staged in LDS via async copies, double-buffered:
//   issue tile t+1 (26 async b32 row-copies), s_wait_asynccnt 26 -> tile t
//   landed (async loads complete in order), compute 8 WMMAs, flip buffers.
// Fragment layouts per CDNA5 ISA 7.12.2 (f32, 16x16x4):
//   A: lane m=(l&15); VGPR0=K{0|2}, VGPR1=K{1|3} (hi lanes -> K+2)
//   B: lane n=(l&15); VGPR0=rows{0|2}, VGPR1=rows{1|3}
//   D: VGPR j, lanes 0-15 -> row j  (rows 0..7 are exactly the kept clusters)
// Row stride 34 keeps paired LDS reads 8B-aligned and bank-conflict free.
// ---------------------------------------------------------------------------
__global__ void __launch_bounds__(32) vlad_wmma_kernel(const float* __restrict__ F,
                                                       const float* __restrict__ SA,
                                                       const float* __restrict__ ASUM,
                                                       const float* __restrict__ CEN,
                                                       float* __restrict__ RES) {
    const int b    = blockIdx.x >> 5;      // batch
    const int ct   = blockIdx.x & 31;      // channel tile
    const int c0   = ct * 16;
    const int lane = threadIdx.x;          // 0..31 (wave32)
    const int n    = lane & 15;            // A row (cluster) == B col (channel)
    const int koff = (lane >> 4) << 1;     // 0 for lanes 0-15, 2 for lanes 16-31

    __shared__ float bsm[2][16][34];       // features: 16 channels x 32 hw
    __shared__ float asm_[2][16][34];      // soft_assign: 10 rows + 6 zero rows

    const float* Fb  = F  + ((size_t)b * VC + c0) * VHW;
    const float* SAb = SA + (size_t)b * VKG * VHW;

    // zero-pad A rows 10..15 once (async copies never touch them)
#pragma unroll
    for (int buf = 0; buf < 2; ++buf)
#pragma unroll
        for (int r = VKG; r < 16; ++r)
            asm_[buf][r][lane] = 0.f;

    auto issue_tile = [&](int hw0, int buf) {
#pragma unroll
        for (int r = 0; r < 16; ++r)
            async_copy_f32(&bsm[buf][r][lane], Fb + r * VHW + hw0 + lane);
#pragma unroll
        for (int r = 0; r < VKG; ++r)
            async_copy_f32(&asm_[buf][r][lane], SAb + r * VHW + hw0 + lane);
    };

    v8f acc = {};
    issue_tile(0, 0);

    int buf = 0;
    for (int hw0 = 0; hw0 < VHW; hw0 += 32, buf ^= 1) {
        if (hw0 + 32 < VHW) {
            issue_tile(hw0 + 32, buf ^ 1);   // prefetch next tile (async)
            wait_asynccnt<26>();             // current tile's 26 copies done
        } else {
            wait_asynccnt<0>();
        }

#pragma unroll
        for (int kk = 0; kk < 32; kk += 4) {
            v2f a, bf;
            a.x  = asm_[buf][n][kk + koff];
            a.y  = asm_[buf][n][kk + koff + 1];
            bf.x = bsm[buf][n][kk + koff];
            bf.y = bsm[buf][n][kk + koff + 1];
            acc = __builtin_amdgcn_wmma_f32_16x16x4_f32(
                /*neg_a=*/false, a, /*neg_b=*/false, bf,
                /*c_mod=*/(short)0, acc, /*reuse_a=*/false, /*reuse_b=*/false);
        }
    }

    // rows 0..7 (kept clusters) live in acc[0..7] of lanes 0-15
    if (lane < 16) {
#pragma unroll
        for (int j = 0; j < VKC; ++j) {
            float r = acc[j] - ASUM[b * VKG + j] * CEN[(size_t)j * VC + c0 + n];
            RES[((size_t)(b * VKC + j)) * VC + c0 + n] = r;
        }
    }
}

// ---------------------------------------------------------------------------
// Kernel 4: L2 normalize each (b,k) row of 512 channels; res and out share
// the same flat layout, so out[i] = res[i] / max(||row||, eps).
// ---------------------------------------------------------------------------
__global__ void __launch_bounds__(256) vlad_norm_kernel(const float* __restrict__ res,
                                                        float* __restrict__ out) {
    int bk = blockIdx.x;                 // 0 .. B*KC-1
    int t  = threadIdx.x;                // 0 .. 255
    const float* row = res + (size_t)bk * VC;
    float v0 = row[t];
    float v1 = row[t + 256];
    __shared__ float red[256];
    red[t] = v0 * v0 + v1 * v1;
    __syncthreads();
#pragma unroll
    for (int off = 128; off > 0; off >>= 1) {
        if (t < off) red[t] += red[t + off];
        __syncthreads();
    }
    float inv = 1.f / fmaxf(sqrtf(red[0]), 1e-12f);
    out[(size_t)bk * VC + t]       = v0 * inv;
    out[(size_t)bk * VC + t + 256] = v1 * inv;
}

// ---------------------------------------------------------------------------
extern "C" void kernel_launch(void* const* d_in, const int* in_sizes, int n_in,
                              void* d_out, int out_size, void* d_ws, size_t ws_size,
                              hipStream_t stream) {
    const float* features = (const float*)d_in[0];   // 32*512*800
    const float* cscore   = (const float*)d_in[1];   // 32*10*800
    const float* centroid = (const float*)d_in[2];   // 10*512
    float*       out      = (float*)d_out;           // 32*4096

    // workspace layout
    float* sa   = (float*)d_ws;                      // 256000 f  soft_assign
    float* asum = sa + (size_t)VB * VKG * VHW;       // 320 f
    float* res  = asum + VB * VKG;                   // 131072 f  un-normalized

    // 1) softmax over batch axis: 8000 (k,hw) positions
    vlad_softmax_kernel<<<(VKG * VHW + 255) / 256, 256, 0, stream>>>(cscore, sa);

    // 2) a_sum[b,k]: one wave per (b,k)
    vlad_asum_kernel<<<VB * VKG, 32, 0, stream>>>(sa, asum);

    // 3) WMMA contraction + centroid subtraction: one wave per (b, ctile)
    vlad_wmma_kernel<<<VB * CTILES, 32, 0, stream>>>(features, sa, asum, centroid, res);

    // 4) L2 normalize per (b,k)
    vlad_norm_kernel<<<VB * VKC, 256, 0, stream>>>(res, out);
}